// MS_63007170232974
// MI455X (gfx1250) — compile-verified
//
#include <hip/hip_runtime.h>

#define C_CH 128
#define KS_ 9
#define PAD_ 4
#define NSPAN 128                 // n positions covered per block (8 waves x 16)
#define STRIP (NSPAN + 2 * PAD_)  // 136 positions incl. halo
#define C_PITCH 132               // channel pitch of transposed strip (132%64==4 -> conflict-free b64)
#define P_ELEMS (STRIP * C_PITCH)        // 17952 floats (71808 B) -> LDS offset 0
#define A_ELEMS (KS_ * 16 * C_CH)        // 18432 floats (73728 B) -> above P

typedef float v2f __attribute__((ext_vector_type(2)));
typedef float v4f __attribute__((ext_vector_type(4)));
typedef float v8f __attribute__((ext_vector_type(8)));

// ---------------------------------------------------------------------------
// Repack w[Cout=128, Cin=128, K=9] into per-lane WMMA A-fragment order:
//   wpk[co_tile(8)][tap(9)][ci_blk(32)][lane(32)][2]
// Lane layout per ISA 7.12.2 (32-bit A-matrix 16x4):
//   lanes 0-15 : M = lane,    K = {0,1}
//   lanes 16-31: M = lane-16, K = {2,3}
// with K mapping to ci = ci_blk*4 + K.
// ---------------------------------------------------------------------------
__global__ __launch_bounds__(256) void spn_repack(const float* __restrict__ w,
                                                  float* __restrict__ wpk) {
  int tid = blockIdx.x * blockDim.x + threadIdx.x;
  const int total = 8 * KS_ * 32 * 32;
  if (tid >= total) return;
  int lane = tid & 31;
  int cb = (tid >> 5) & 31;
  int t = (tid >> 10) % KS_;
  int ct = tid / (KS_ * 32 * 32);
  int m = lane & 15;
  int kb = (lane >> 4) << 1;  // 0 or 2
  int co = ct * 16 + m;
  int ci = cb * 4 + kb;
  wpk[2 * tid + 0] = w[(co * C_CH + ci + 0) * KS_ + t];
  wpk[2 * tid + 1] = w[(co * C_CH + ci + 1) * KS_ + t];
}

// ---------------------------------------------------------------------------
// One scan step: buf[:, :, row, :] = src[:, :, row, :] +
//                relu( conv1d_C->C_K9( buf[:, :, prow, :] ) )
// Tensor layout [B, C, S, L]; conv along L (contiguous).
//
// Block = one (Cout tile, 128-wide n group); 8 waves share the Cout tile and
// each owns a 16-wide N tile. One LDS pool with P (transposed prev-row strip)
// at offset 0 so ALL B-fragment inner offsets fit the 16-bit DS immediate;
// all 9 A panels live above it (per-tap base pointer). Single barrier, then
// 288 immediate-offset ds_load_b64 pairs feeding V_WMMA_F32_16X16X4_F32.
// ---------------------------------------------------------------------------
__global__ __launch_bounds__(256) void spn_step(float* __restrict__ buf,
                                                const float* __restrict__ src,
                                                const float* __restrict__ wpk,
                                                int row, int prow, int S, int L) {
  const int lane = threadIdx.x & 31;
  const int wave = threadIdx.x >> 5;
  const int ct = blockIdx.x & 7;        // Cout tile (8 tiles of 16)
  const int g = blockIdx.x >> 3;        // n-group index
  const int groupsPerB = L / NSPAN;     // 2 (vertical) or 1 (horizontal)
  const int b = g / groupsPerB;
  const int l0g = (g - b * groupsPerB) * NSPAN;

  // Single pool => P is guaranteed at LDS offset 0, Abuf above it.
  __shared__ float ldsPool[P_ELEMS + A_ELEMS];
  float* Pp = ldsPool;            // transposed strip [pos][ci], pitch 132
  float* Ab = ldsPool + P_ELEMS;  // 9 A panels, fragment order

  const size_t chanStride = (size_t)S * L;

  // Stage previous-row strip, transposed: P[p][ci] for p in [0,136), ci in [0,128).
  // Global reads coalesced along l; LDS write stride 132 dwords (bank-staggered).
  const float* prevBase = buf + (size_t)b * C_CH * chanStride + (size_t)prow * L;
  for (int idx = threadIdx.x; idx < C_CH * STRIP; idx += 256) {
    int ci = idx / STRIP;
    int p = idx - ci * STRIP;
    int l = l0g - PAD_ + p;
    float v = 0.0f;
    if (l >= 0 && l < L) v = prevBase[(size_t)ci * chanStride + l];
    Pp[p * C_PITCH + ci] = v;
  }

  // Stage all 9 A panels for this Cout tile: 18432 contiguous floats.
  {
    const v4f* s4 = (const v4f*)(wpk + (size_t)ct * A_ELEMS);
    v4f* d4 = (v4f*)Ab;
    for (int i = threadIdx.x; i < A_ELEMS / 4; i += 256) d4[i] = s4[i];
  }
  __syncthreads();

  const int ncol = lane & 15;       // N column for B/C/D fragments
  const int kb = (lane >> 4) << 1;  // K sub-index 0 or 2 for this half-wave
  const int nbase = wave * 16;      // this wave's N tile inside the group

  v8f acc = {};
  for (int t = 0; t < KS_; ++t) {
    // Per-tap base pointers; inner offsets are compile-time constants that fit
    // the DS immediate field (B: cb*16 B from offset<71808; A: cb*256 B).
    const float* aB = Ab + t * (32 * 32 * 2) + lane * 2;
    const float* bB = Pp + (nbase + ncol + t) * C_PITCH + kb;
#pragma unroll
    for (int cb = 0; cb < 32; ++cb) {
      v2f a = *(const v2f*)(aB + cb * 64);
      v2f bf = *(const v2f*)(bB + cb * 4);
      acc = __builtin_amdgcn_wmma_f32_16x16x4_f32(false, a, false, bf,
                                                  (short)0, acc, false, false);
    }
  }

  // Epilogue: C/D layout (ISA 7.12.2): lanes 0-15 -> M = v, lanes 16-31 -> M = 8+v.
  const int mbase = (lane >> 4) << 3;
  size_t base = ((size_t)b * C_CH + ct * 16 + mbase) * chanStride +
                (size_t)row * L + l0g + nbase + ncol;
  const float* sp = src + base;
  float* op = buf + base;
#pragma unroll
  for (int v = 0; v < 8; ++v) {
    float r = acc[v];
    r = r > 0.0f ? r : 0.0f;
    op[(size_t)v * chanStride] = sp[(size_t)v * chanStride] + r;
  }
}

// Copy scan-row 0 of pass 1 (out row0 = x row0).
__global__ __launch_bounds__(256) void spn_copy_row0(float* __restrict__ out,
                                                     const float* __restrict__ x,
                                                     int S, int L, int n) {
  int tid = blockIdx.x * blockDim.x + threadIdx.x;
  if (tid >= n) return;
  int l = tid % L;
  int bc = tid / L;
  size_t idx = (size_t)bc * S * L + l;
  out[idx] = x[idx];
}

// Transpose last two dims: in[BC][S][L] -> out[BC][L][S].
__global__ __launch_bounds__(256) void spn_transpose(const float* __restrict__ in,
                                                     float* __restrict__ outp,
                                                     int S, int L) {
  __shared__ float tile[32][33];
  const int bc = blockIdx.z;
  const size_t plane = (size_t)S * L;
  const float* ip = in + (size_t)bc * plane;
  float* op = outp + (size_t)bc * plane;
  const int l0 = blockIdx.x * 32, s0 = blockIdx.y * 32;
  const int tx = threadIdx.x & 31, ty = threadIdx.x >> 5;
#pragma unroll
  for (int r = 0; r < 32; r += 8)
    tile[ty + r][tx] = ip[(size_t)(s0 + ty + r) * L + (l0 + tx)];
  __syncthreads();
#pragma unroll
  for (int r = 0; r < 32; r += 8)
    op[(size_t)(l0 + ty + r) * S + (s0 + tx)] = tile[tx][ty + r];
}

extern "C" void kernel_launch(void* const* d_in, const int* in_sizes, int n_in,
                              void* d_out, int out_size, void* d_ws, size_t ws_size,
                              hipStream_t stream) {
  const float* x = (const float*)d_in[0];
  const float* w_ud = (const float*)d_in[1];
  const float* w_du = (const float*)d_in[2];
  const float* w_lr = (const float*)d_in[3];
  const float* w_rl = (const float*)d_in[4];
  float* out = (float*)d_out;

  const int H = 128, W = 256;
  const int B = in_sizes[0] / (C_CH * H * W);

  // Workspace map: [0, B*C*H*W) transposed tensor, then 4 packed weight blocks.
  float* T2 = (float*)d_ws;
  float* wpk = T2 + (size_t)B * C_CH * H * W;
  const size_t WPK_ELEMS = (size_t)8 * KS_ * 32 * 32 * 2;

  dim3 blk(256);
  int rpBlocks = (8 * KS_ * 32 * 32 + 255) / 256;
  spn_repack<<<rpBlocks, blk, 0, stream>>>(w_ud, wpk + 0 * WPK_ELEMS);
  spn_repack<<<rpBlocks, blk, 0, stream>>>(w_du, wpk + 1 * WPK_ELEMS);
  spn_repack<<<rpBlocks, blk, 0, stream>>>(w_lr, wpk + 2 * WPK_ELEMS);
  spn_repack<<<rpBlocks, blk, 0, stream>>>(w_rl, wpk + 3 * WPK_ELEMS);

  // Pass 1: up -> down (scan H, conv along W), src = x, dst = out.
  int n0 = B * C_CH * W;
  spn_copy_row0<<<(n0 + 255) / 256, blk, 0, stream>>>(out, x, H, W, n0);
  int gV = 8 * (B * W / NSPAN);  // 8 Cout tiles x n-groups
  for (int i = 1; i < H; ++i)
    spn_step<<<gV, blk, 0, stream>>>(out, x, wpk + 0 * WPK_ELEMS, i, i - 1, H, W);

  // Pass 2: down -> up, in-place on out.
  for (int i = H - 2; i >= 0; --i)
    spn_step<<<gV, blk, 0, stream>>>(out, out, wpk + 1 * WPK_ELEMS, i, i + 1, H, W);

  // Transpose [B,C,H,W] -> [B,C,W,H] so horizontal scans have a contiguous conv axis.
  dim3 tg1(W / 32, H / 32, B * C_CH);
  spn_transpose<<<tg1, blk, 0, stream>>>(out, T2, H, W);

  // Pass 3: left -> right (scan W, conv along H), in-place on T2.
  int gH = 8 * (B * H / NSPAN);
  for (int i = 1; i < W; ++i)
    spn_step<<<gH, blk, 0, stream>>>(T2, T2, wpk + 2 * WPK_ELEMS, i, i - 1, W, H);

  // Pass 4: right -> left.
  for (int i = W - 2; i >= 0; --i)
    spn_step<<<gH, blk, 0, stream>>>(T2, T2, wpk + 3 * WPK_ELEMS, i, i + 1, W, H);

  // Transpose back [B,C,W,H] -> [B,C,H,W].
  dim3 tg2(H / 32, W / 32, B * C_CH);
  spn_transpose<<<tg2, blk, 0, stream>>>(T2, out, W, H);
}